// gridded_nufft_75522704933327
// MI455X (gfx1250) — compile-verified
//
#include <hip/hip_runtime.h>
#include <math.h>

typedef __attribute__((ext_vector_type(2))) float v2f;
typedef __attribute__((ext_vector_type(8))) float v8f;

#define GS   480     // oversampled grid size
#define IS   384     // image size
#define NCH  16
#define NPTS 500000

// ---------------------------------------------------------------------------
// Build centered-DFT matrix restricted to the padded input support:
//   F[k, n] = exp(-2*pi*i*(k-240)*(n+48-240)/480) / sqrt(480),  k<480, n<384
// and its transpose FT[n, k] for the stage-B (right-multiply) GEMM.
// Integer phase reduction mod 480 keeps sincos arguments small & exact.
// ---------------------------------------------------------------------------
__global__ void build_dft(float* __restrict__ fr, float* __restrict__ fi,
                          float* __restrict__ ftr, float* __restrict__ fti) {
  int idx = blockIdx.x * blockDim.x + threadIdx.x;
  if (idx >= GS * IS) return;
  int k = idx / IS;
  int n = idx - k * IS;
  long long p = (long long)(k - GS / 2) * (long long)(n + 48 - GS / 2);
  int m = (int)(p % GS);
  if (m < 0) m += GS;
  float ph = -6.283185307179586f * ((float)m / (float)GS);
  float s, c;
  __sincosf(ph, &s, &c);
  const float scale = 0.04564354645876384f;  // 1/sqrt(480) (ortho norm per dim)
  float vr = c * scale, vi = s * scale;
  fr[k * IS + n]  = vr;
  fi[k * IS + n]  = vi;
  ftr[n * GS + k] = vr;
  fti[n * GS + k] = vi;
}

// ---------------------------------------------------------------------------
// Complex GEMM via V_WMMA_F32_16X16X4_F32, register-blocked 2x along N.
//   C[ch] = A[ch] (Mt*16 x K) * B[ch] (K x Nt2*32), complex (split re/im).
// One wave32 computes a 16x32 complex output tile: 8 f32 WMMA chains
// (RR/II/RI/IR for each of 2 N-subtiles) sharing one A operand load pair.
// Cr = RR - II, Ci = RI + IR combined in VALU at the end.
// Output supports an element stride (cElem) so stage B can interleave channels.
// ---------------------------------------------------------------------------
__global__ void __launch_bounds__(128)
cgemm_wmma(const float* __restrict__ Ar, const float* __restrict__ Ai,
           int lda, long long aCh,
           const float* __restrict__ Br, const float* __restrict__ Bi,
           int ldb, long long bCh,
           float* __restrict__ Cr, float* __restrict__ Ci,
           int ldc, int cElem, long long cCh,
           int Mt, int Nt2, int K) {
  const int wave = threadIdx.x >> 5;
  const int lane = threadIdx.x & 31;
  const int tile = blockIdx.x * 4 + wave;
  if (tile >= Mt * Nt2) return;             // wave-uniform guard (EXEC stays ~0)
  const int ch = blockIdx.y;
  const int mt = tile / Nt2;
  const int m0 = mt << 4;
  const int n0 = (tile - mt * Nt2) << 5;    // 32-wide N block

  const float* ar = Ar + aCh * ch;
  const float* ai = Ai + aCh * ch;
  const float* br = Br + bCh * ch;
  const float* bi = Bi + bCh * ch;

  // ISA A layout (16x4 f32): lanes0-15 -> M, VGPR{0,1}=K{0,1}; lanes16-31 -> K{2,3}
  const int am   = lane & 15;
  const int koff = (lane >> 4) << 1;        // 0 or 2
  const float* arow_r = ar + (size_t)(m0 + am) * lda + koff;
  const float* arow_i = ai + (size_t)(m0 + am) * lda + koff;
  // B layout (4x16 f32) mirrors A: lane -> N, VGPR{0,1} = rows K{koff, koff+1}
  const float* b_r0 = br + (size_t)koff * ldb + (n0 + am);
  const float* b_i0 = bi + (size_t)koff * ldb + (n0 + am);
  const float* b_r1 = b_r0 + 16;
  const float* b_i1 = b_i0 + 16;

  v8f rr0 = {}, ii0 = {}, ri0 = {}, ir0 = {};
  v8f rr1 = {}, ii1 = {}, ri1 = {}, ir1 = {};

  const size_t bStep = (size_t)4 * ldb;

  for (int k = 0; k < K; k += 4) {
    // A operands: one 8B (b64) load each for re/im, shared by both N-subtiles
    v2f a_r = *reinterpret_cast<const v2f*>(arow_r);
    v2f a_i = *reinterpret_cast<const v2f*>(arow_i);
    v2f br0v, bi0v, br1v, bi1v;
    br0v[0] = b_r0[0];   br0v[1] = b_r0[ldb];
    bi0v[0] = b_i0[0];   bi0v[1] = b_i0[ldb];
    br1v[0] = b_r1[0];   br1v[1] = b_r1[ldb];
    bi1v[0] = b_i1[0];   bi1v[1] = b_i1[ldb];

    rr0 = __builtin_amdgcn_wmma_f32_16x16x4_f32(false, a_r, false, br0v, (short)0, rr0, false, false);
    ii0 = __builtin_amdgcn_wmma_f32_16x16x4_f32(false, a_i, false, bi0v, (short)0, ii0, false, false);
    ri0 = __builtin_amdgcn_wmma_f32_16x16x4_f32(false, a_r, false, bi0v, (short)0, ri0, false, false);
    ir0 = __builtin_amdgcn_wmma_f32_16x16x4_f32(false, a_i, false, br0v, (short)0, ir0, false, false);
    rr1 = __builtin_amdgcn_wmma_f32_16x16x4_f32(false, a_r, false, br1v, (short)0, rr1, false, false);
    ii1 = __builtin_amdgcn_wmma_f32_16x16x4_f32(false, a_i, false, bi1v, (short)0, ii1, false, false);
    ri1 = __builtin_amdgcn_wmma_f32_16x16x4_f32(false, a_r, false, bi1v, (short)0, ri1, false, false);
    ir1 = __builtin_amdgcn_wmma_f32_16x16x4_f32(false, a_i, false, br1v, (short)0, ir1, false, false);

    arow_r += 4;  arow_i += 4;
    b_r0 += bStep; b_i0 += bStep; b_r1 += bStep; b_i1 += bStep;
  }

  v8f cr0 = rr0 - ii0, ci0 = ri0 + ir0;
  v8f cr1 = rr1 - ii1, ci1 = ri1 + ir1;

  // C/D layout: VGPR r -> M=r (lanes0-15) or M=r+8 (lanes16-31), N=lane%16
  const int mBase = m0 + ((lane >> 4) << 3);
  const int nOut  = n0 + (lane & 15);
  float* outR = Cr + cCh * ch;
  float* outI = Ci + cCh * ch;
#pragma unroll
  for (int r = 0; r < 8; ++r) {
    size_t off0 = ((size_t)(mBase + r) * ldc + nOut) * (size_t)cElem;
    size_t off1 = ((size_t)(mBase + r) * ldc + nOut + 16) * (size_t)cElem;
    outR[off0] = cr0[r];
    outI[off0] = ci0[r];
    outR[off1] = cr1[r];
    outI[off1] = ci1[r];
  }
}

// ---------------------------------------------------------------------------
// Scattered gather: per point, round trajectory to grid indices (clip + rint,
// matching jnp round-half-even), read 16 channels (contiguous 64B re + 64B im
// thanks to channel-interleaved grid), write coalesced along K.
// out layout: (2, 1, 16, NPTS)
// ---------------------------------------------------------------------------
__global__ void gather_pts(const float* __restrict__ trj,
                           const float* __restrict__ gr,
                           const float* __restrict__ gi,
                           float* __restrict__ out, int npts) {
  int k = blockIdx.x * blockDim.x + threadIdx.x;
  if (k >= npts) return;
  float tx = trj[2 * k + 0];
  float ty = trj[2 * k + 1];
  float fx = fminf(fmaxf(tx * 1.25f + 240.0f, 0.0f), 479.0f);
  float fy = fminf(fmaxf(ty * 1.25f + 240.0f, 0.0f), 479.0f);
  int ix = (int)rintf(fx);
  int iy = (int)rintf(fy);
  size_t base = ((size_t)ix * GS + iy) * NCH;
#pragma unroll
  for (int c = 0; c < NCH; ++c) {
    out[(size_t)c * npts + k]          = gr[base + c];
    out[(size_t)(NCH + c) * npts + k]  = gi[base + c];
  }
}

// ---------------------------------------------------------------------------
extern "C" void kernel_launch(void* const* d_in, const int* in_sizes, int n_in,
                              void* d_out, int out_size, void* d_ws, size_t ws_size,
                              hipStream_t stream) {
  const float* img_r = (const float*)d_in[0];  // (1,16,384,384)
  const float* img_i = (const float*)d_in[1];  // (1,16,384,384)
  const float* trj   = (const float*)d_in[2];  // (1,500000,2)
  float* out = (float*)d_out;                  // (2,1,16,500000)

  float* ws = (float*)d_ws;
  size_t o = 0;
  float* Fr  = ws + o; o += (size_t)GS * IS;          // 480x384 DFT re
  float* Fi  = ws + o; o += (size_t)GS * IS;          // 480x384 DFT im
  float* FTr = ws + o; o += (size_t)IS * GS;          // 384x480 transpose re
  float* FTi = ws + o; o += (size_t)IS * GS;          // 384x480 transpose im
  float* Tr  = ws + o; o += (size_t)NCH * GS * IS;    // stage-A tmp re
  float* Ti  = ws + o; o += (size_t)NCH * GS * IS;    // stage-A tmp im
  float* Gr  = ws + o; o += (size_t)GS * GS * NCH;    // k-space grid re [kx][ky][c]
  float* Gi  = ws + o; o += (size_t)GS * GS * NCH;    // k-space grid im

  // 1) DFT matrices
  {
    int n = GS * IS;
    build_dft<<<(n + 255) / 256, 256, 0, stream>>>(Fr, Fi, FTr, FTi);
  }
  // 2) Stage A: Tmp[c] (480x384) = F (480x384) * img[c] (384x384)
  //    30 M-tiles x 12 (32-wide) N-blocks = 360 wave-tiles -> 90 blocks
  {
    dim3 grid(90, NCH);
    cgemm_wmma<<<grid, 128, 0, stream>>>(
        Fr, Fi, IS, 0LL,
        img_r, img_i, IS, (long long)IS * IS,
        Tr, Ti, IS, 1, (long long)GS * IS,
        30, 12, IS);
  }
  // 3) Stage B: Grid[c] (480x480) = Tmp[c] (480x384) * F^T (384x480),
  //    channel-interleaved output. 30 x 15 = 450 wave-tiles -> 113 blocks.
  {
    dim3 grid(113, NCH);
    cgemm_wmma<<<grid, 128, 0, stream>>>(
        Tr, Ti, IS, (long long)GS * IS,
        FTr, FTi, GS, 0LL,
        Gr, Gi, GS, NCH, 1LL,
        30, 15, IS);
  }
  // 4) Scattered gather of 500k trajectory points
  gather_pts<<<(NPTS + 255) / 256, 256, 0, stream>>>(trj, Gr, Gi, out, NPTS);
}